// DetectionLoss_25512105739113
// MI455X (gfx1250) — compile-verified
//
#include <hip/hip_runtime.h>
#include <hip/hip_bf16.h>

// ---------------------------------------------------------------------------
// Detection loss (SSD-style) for gfx1250 / MI455X, wave32.
// ---------------------------------------------------------------------------

#define EPSF 1e-6f

#define NLVL 3
#define BATCH 32
#define NGT 20
#define KEY_PER_B 64512              // 49152 + 12288 + 3072
#define NB1 (192 * 32)               // level-1 blocks (49152/256 * 32)
#define NB2 (48 * 32)
#define NB3 (12 * 32)
#define NB_TOT (NB1 + NB2 + NB3)     // 8064
#define MINING_LDS_BYTES (49152 * 4) // largest level slice staged in LDS (192 KB <= 320 KB/WGP)

typedef float v2f __attribute__((ext_vector_type(2)));
typedef float v8f __attribute__((ext_vector_type(8)));

// Full-wave (32 lane) sum via V_WMMA_F32_16X16X4_F32.
// A[m][k]: lane m holds (a.x=K0, a.y=K1), lane m+16 holds (K2, K3).
// With a.y = 0 and B = ones(4x16):  D[m][n] = v[m] + v[m+16]  (all n equal).
// C/D layout: lane L<16 holds D[0..7][L] -> 8-sum = S_low; lane L>=16 -> S_high.
// S_low + S_high (one shfl_xor) = full 32-lane sum on every lane.
// EXEC must be all ones at the call site (no divergence; grids have no tail).
__device__ __forceinline__ float wmma_wave_sum(float v) {
  v2f a; a[0] = v;    a[1] = 0.0f;
  v2f b; b[0] = 1.0f; b[1] = 1.0f;
  v8f c = {};
  c = __builtin_amdgcn_wmma_f32_16x16x4_f32(false, a, false, b,
                                            (short)0, c, false, false);
  float s = ((c[0] + c[1]) + (c[2] + c[3])) + ((c[4] + c[5]) + (c[6] + c[7]));
  s += __shfl_xor(s, 16, 32);
  return s;
}

__device__ __forceinline__ float smooth_l1(float d) {
  float ad = fabsf(d);
  return (ad < 1.0f) ? 0.5f * d * d : ad - 0.5f;
}

// ---------------------------------------------------------------------------
__global__ void det_zero_kernel(unsigned* __restrict__ cnts) {
  int t = blockIdx.x * blockDim.x + threadIdx.x;
  if (t < 192) cnts[t] = 0u;   // pos_cnt[96] + neg_cnt[96]
}

// ---------------------------------------------------------------------------
// One thread per anchor. grid = (N/256, 32). N divisible by 256 -> no tail.
__global__ void det_anchor_kernel(const float* __restrict__ pred,
                                  const float* __restrict__ anchors,
                                  const float* __restrict__ gt_boxes,
                                  const int*   __restrict__ gt_labels,
                                  unsigned*    __restrict__ keys,
                                  unsigned*    __restrict__ pos_cnt,
                                  unsigned*    __restrict__ neg_cnt,
                                  float*       __restrict__ partial,
                                  int H, int lvl, int key_off, int block_off) {
  __shared__ float s_gt[NGT * 4];
  __shared__ int   s_lab[NGT];
  __shared__ float s_red[3][8];

  const int tid = threadIdx.x;
  const int b   = blockIdx.y;
  const int i   = blockIdx.x * 256 + tid;

  if (tid < NGT * 4)                  s_gt[tid]       = gt_boxes[b * NGT * 4 + tid];
  if (tid >= 128 && tid < 128 + NGT)  s_lab[tid - 128] = gt_labels[b * NGT + (tid - 128)];
  __syncthreads();

  // ---- anchor box ----
  const float ax1 = anchors[i * 4 + 0], ay1 = anchors[i * 4 + 1];
  const float ax2 = anchors[i * 4 + 2], ay2 = anchors[i * 4 + 3];
  const float aarea = (ax2 - ax1) * (ay2 - ay1);

  // ---- best IoU over 20 GT (first-max wins, as jnp.argmax) ----
  float best = -1.0f; int bi = 0;
  #pragma unroll 4
  for (int m = 0; m < NGT; ++m) {
    float gx1 = s_gt[m * 4 + 0], gy1 = s_gt[m * 4 + 1];
    float gx2 = s_gt[m * 4 + 2], gy2 = s_gt[m * 4 + 3];
    float iw = fmaxf(fminf(ax2, gx2) - fmaxf(ax1, gx1), 0.0f);
    float ih = fmaxf(fminf(ay2, gy2) - fmaxf(ay1, gy1), 0.0f);
    float inter = iw * ih;
    float garea = (gx2 - gx1) * (gy2 - gy1);
    float iou = inter / (aarea + garea - inter + EPSF);
    if (iou > best) { best = iou; bi = m; }
  }
  const bool pos = best >= 0.5f;
  const bool neg = best < 0.3f;

  // ---- gather 8 channels: pred[b, a*8+c, h, w]; anchor i = (h*W+w)*3 + a ----
  const size_t HW = (size_t)H * (size_t)H;
  const int a  = i % 3;
  const int hw = i / 3;
  const float* pb = pred + ((size_t)(b * 24 + a * 8)) * HW + (size_t)hw;
  const float p0 = pb[0 * HW], p1 = pb[1 * HW], p2 = pb[2 * HW], p3 = pb[3 * HW];
  const float px = pb[4 * HW];
  const float l0 = pb[5 * HW], l1 = pb[6 * HW], l2 = pb[7 * HW];

  // ---- objectness BCE-with-logits ----
  const float t    = pos ? 1.0f : 0.0f;
  const float objl = fmaxf(px, 0.0f) - px * t + log1pf(expf(-fabsf(px)));

  // radix key: 0 => not negative; else monotone bits(objl)+1 (objl >= 0)
  keys[(size_t)b * KEY_PER_B + key_off + i] = neg ? (__float_as_uint(objl) + 1u) : 0u;

  // ---- class CE (only counted when pos) ----
  int tgt = s_lab[bi]; tgt = (tgt < 0) ? 0 : ((tgt > 2) ? 2 : tgt);
  const float mx  = fmaxf(l0, fmaxf(l1, l2));
  const float lse = mx + logf(expf(l0 - mx) + expf(l1 - mx) + expf(l2 - mx));
  const float ce  = lse - ((tgt == 0) ? l0 : ((tgt == 1) ? l1 : l2));

  // ---- smooth-L1 on encoded box deltas ----
  const float gx1 = s_gt[bi * 4 + 0], gy1 = s_gt[bi * 4 + 1];
  const float gx2 = s_gt[bi * 4 + 2], gy2 = s_gt[bi * 4 + 3];
  const float gw = fmaxf(gx2 - gx1, EPSF), gh = fmaxf(gy2 - gy1, EPSF);
  const float aw = fmaxf(ax2 - ax1, EPSF), ah = fmaxf(ay2 - ay1, EPSF);
  const float gcx = gx1 + 0.5f * gw, gcy = gy1 + 0.5f * gh;
  const float acx = ax1 + 0.5f * aw, acy = ay1 + 0.5f * ah;
  const float tx = (gcx - acx) / (aw + EPSF);
  const float ty = (gcy - acy) / (ah + EPSF);
  const float tw = logf((gw + EPSF) / (aw + EPSF));
  const float th = logf((gh + EPSF) / (ah + EPSF));
  const float sl1 = smooth_l1(p0 - tx) + smooth_l1(p1 - ty) +
                    smooth_l1(p2 - tw) + smooth_l1(p3 - th);

  const float obj_v = pos ? objl : 0.0f;
  const float cls_v = pos ? ce   : 0.0f;
  const float loc_v = pos ? sl1  : 0.0f;

  // ---- exact integer counts per (image,level) via ballot ----
  const unsigned lane = tid & 31u;
  const int wave = tid >> 5;
  const int il = lvl * 32 + b;
  unsigned long long bp = __ballot((int)pos);
  unsigned long long bn = __ballot((int)neg);
  if (lane == 0u) {
    unsigned cp = (unsigned)__popcll(bp);
    unsigned cn = (unsigned)__popcll(bn);
    if (cp) atomicAdd(&pos_cnt[il], cp);
    if (cn) atomicAdd(&neg_cnt[il], cn);
  }

  // ---- wave-level f32 sums via WMMA (all 32 lanes active) ----
  const float so = wmma_wave_sum(obj_v);
  const float sc = wmma_wave_sum(cls_v);
  const float sl = wmma_wave_sum(loc_v);
  if (lane == 0u) { s_red[0][wave] = so; s_red[1][wave] = sc; s_red[2][wave] = sl; }
  __syncthreads();
  if (tid == 0) {   // deterministic per-block combine, no float atomics
    float t0 = 0.f, t1 = 0.f, t2 = 0.f;
    #pragma unroll
    for (int w2 = 0; w2 < 8; ++w2) {
      t0 += s_red[0][w2]; t1 += s_red[1][w2]; t2 += s_red[2][w2];
    }
    const int blk = block_off + blockIdx.y * gridDim.x + blockIdx.x;
    partial[blk]              = t0;
    partial[NB_TOT + blk]     = t1;
    partial[2 * NB_TOT + blk] = t2;
  }
}

// ---------------------------------------------------------------------------
// Hard-negative mining: one block per (image,level).
// Stage the whole key slice into LDS ONCE with async global->LDS copies
// (CDNA5 ASYNCcnt path), then run all 8 MSB-first 4-bit radix-select passes
// out of LDS: 8x less L2 traffic than re-reading global per pass.
__global__ void det_mining_kernel(const unsigned* __restrict__ keys,
                                  const unsigned* __restrict__ pos_cnt,
                                  const unsigned* __restrict__ neg_cnt,
                                  float*    __restrict__ mined_sum,
                                  unsigned* __restrict__ mined_cnt) {
  extern __shared__ unsigned s_keys[];   // up to 49152 words (192 KB)

  const int il  = blockIdx.x;      // lvl*32 + b
  const int lvl = il >> 5;
  const int b   = il & 31;
  const int Ns[NLVL]   = {49152, 12288, 3072};
  const int offs[NLVL] = {0, 49152, 61440};
  const int N = Ns[lvl];
  const unsigned* kb = keys + (size_t)b * KEY_PER_B + offs[lvl];
  const int tid = threadIdx.x;

  const unsigned p_cnt = pos_cnt[il];
  const unsigned n_cnt = neg_cnt[il];
  const unsigned k     = 3u * (p_cnt > 1u ? p_cnt : 1u);
  const unsigned k_eff = (k < n_cnt) ? k : n_cnt;
  if (k_eff == 0u) {
    if (tid == 0) { mined_sum[il] = 0.0f; mined_cnt[il] = 0u; }
    return;
  }

  // ---- async stage: 16 B per lane per issue; N is a multiple of 1024 so the
  // trip count is uniform and EXEC stays all-ones. Each wave waits for its own
  // ASYNCcnt, then the barrier makes the staged data visible block-wide.
  {
    const unsigned lds_base = (unsigned)(unsigned long long)(&s_keys[0]);
    for (int j4 = tid * 4; j4 < N; j4 += 1024) {
      unsigned lds_off = lds_base + (unsigned)j4 * 4u;
      unsigned goff    = (unsigned)j4 * 4u;
      asm volatile("global_load_async_to_lds_b128 %0, %1, %2"
                   :: "v"(lds_off), "v"(goff), "s"(kb)
                   : "memory");
    }
    asm volatile("s_wait_asynccnt 0" ::: "memory");
  }
  __syncthreads();

  __shared__ unsigned s_cnt[16];
  __shared__ float    s_sum[16];
  __shared__ unsigned sh_prefix, sh_remaining;
  __shared__ float    sh_selsum;
  if (tid == 0) { sh_prefix = 0u; sh_remaining = k_eff; sh_selsum = 0.0f; }

  for (int shift = 28; shift >= 0; shift -= 4) {
    __syncthreads();
    if (tid < 16) { s_cnt[tid] = 0u; s_sum[tid] = 0.0f; }
    __syncthreads();
    const unsigned prefix  = sh_prefix;
    const unsigned mask_hi = (shift == 28) ? 0u : (0xFFFFFFFFu << (shift + 4));
    for (int j = tid; j < N; j += 256) {          // conflict-free LDS stream
      const unsigned key = s_keys[j];
      if (key != 0u && (key & mask_hi) == (prefix & mask_hi)) {
        const unsigned d = (key >> shift) & 0xFu;
        atomicAdd(&s_cnt[d], 1u);
        atomicAdd(&s_sum[d], __uint_as_float(key - 1u));
      }
    }
    __syncthreads();
    if (tid == 0) {
      unsigned rem = sh_remaining;
      float    ss  = sh_selsum;
      int v = 15;
      for (; v > 0; --v) {
        const unsigned c = s_cnt[v];
        if (c >= rem) break;          // k-th element has this digit
        rem -= c; ss += s_sum[v];     // strictly-greater digits fully selected
      }
      sh_remaining = rem;
      sh_selsum    = ss;
      sh_prefix    = prefix | ((unsigned)v << shift);
    }
  }
  __syncthreads();
  if (tid == 0) {
    const float tie_val = __uint_as_float(sh_prefix - 1u); // exact k-th value
    mined_sum[il] = sh_selsum + (float)sh_remaining * tie_val;
    mined_cnt[il] = k_eff;
  }
}

// ---------------------------------------------------------------------------
__global__ void det_finalize_kernel(const float*    __restrict__ partial,
                                    const float*    __restrict__ mined_sum,
                                    const unsigned* __restrict__ pos_cnt,
                                    const unsigned* __restrict__ mined_cnt,
                                    float* __restrict__ out) {
  __shared__ float red[3][256];
  const int tid = threadIdx.x;
  float o = 0.f, c = 0.f, l = 0.f;
  for (int j = tid; j < NB_TOT; j += 256) {
    __builtin_prefetch(&partial[j + 256], 0, 0);   // global_prefetch_b8
    o += partial[j];
    c += partial[NB_TOT + j];
    l += partial[2 * NB_TOT + j];
  }
  if (tid < 96) o += mined_sum[tid];
  red[0][tid] = o; red[1][tid] = c; red[2][tid] = l;
  __syncthreads();
  for (int s2 = 128; s2 > 0; s2 >>= 1) {
    if (tid < s2) {
      red[0][tid] += red[0][tid + s2];
      red[1][tid] += red[1][tid + s2];
      red[2][tid] += red[2][tid + s2];
    }
    __syncthreads();
  }
  if (tid == 0) {
    unsigned npos = 0u, den_x = 0u;
    for (int j = 0; j < 96; ++j) { npos += pos_cnt[j]; den_x += mined_cnt[j]; }
    const float obj_sum = red[0][0], cls_sum = red[1][0], loc_sum = red[2][0];
    const unsigned den_u = npos + den_x;
    const float pos_norm = (float)(npos  > 1u ? npos  : 1u);
    const float obj_norm = (float)(den_u > 1u ? den_u : 1u);
    const float lo = obj_sum / obj_norm;
    const float lc = cls_sum / pos_norm;
    const float ll = loc_sum / pos_norm;
    out[0] = lo; out[1] = lc; out[2] = ll; out[3] = lo + lc + 2.0f * ll;
  }
}

// ---------------------------------------------------------------------------
extern "C" void kernel_launch(void* const* d_in, const int* in_sizes, int n_in,
                              void* d_out, int out_size, void* d_ws, size_t ws_size,
                              hipStream_t stream) {
  const float* pred1 = (const float*)d_in[0];
  const float* anc1  = (const float*)d_in[1];
  const float* pred2 = (const float*)d_in[2];
  const float* anc2  = (const float*)d_in[3];
  const float* pred3 = (const float*)d_in[4];
  const float* anc3  = (const float*)d_in[5];
  const float* gtb   = (const float*)d_in[6];
  const int*   gtl   = (const int*)d_in[7];
  float* out = (float*)d_out;

  // Workspace layout (u32 units):
  //   [0,96)    pos_cnt   [96,192) neg_cnt   [192,288) mined_cnt   pad->512
  //   [512, 512 + 32*64512)  radix keys
  //   then floats: partial[3*NB_TOT], mined_sum[96]      (~8.4 MB total)
  unsigned* ws_u      = (unsigned*)d_ws;
  unsigned* pos_cnt   = ws_u;
  unsigned* neg_cnt   = ws_u + 96;
  unsigned* mined_cnt = ws_u + 192;
  unsigned* keys      = ws_u + 512;
  float*    fbase     = (float*)(ws_u + 512 + (size_t)BATCH * KEY_PER_B);
  float*    partial   = fbase;
  float*    mined_sum = fbase + 3 * NB_TOT;

  // Allow the 192 KB dynamic-LDS staging buffer (CDNA5: 320 KB per WGP).
  (void)hipFuncSetAttribute((const void*)det_mining_kernel,
                            hipFuncAttributeMaxDynamicSharedMemorySize,
                            MINING_LDS_BYTES + 1024);

  det_zero_kernel<<<1, 256, 0, stream>>>(ws_u);

  det_anchor_kernel<<<dim3(192, 32), 256, 0, stream>>>(
      pred1, anc1, gtb, gtl, keys, pos_cnt, neg_cnt, partial,
      128, 0, 0, 0);
  det_anchor_kernel<<<dim3(48, 32), 256, 0, stream>>>(
      pred2, anc2, gtb, gtl, keys, pos_cnt, neg_cnt, partial,
      64, 1, 49152, NB1);
  det_anchor_kernel<<<dim3(12, 32), 256, 0, stream>>>(
      pred3, anc3, gtb, gtl, keys, pos_cnt, neg_cnt, partial,
      32, 2, 61440, NB1 + NB2);

  det_mining_kernel<<<96, 256, MINING_LDS_BYTES, stream>>>(
      keys, pos_cnt, neg_cnt, mined_sum, mined_cnt);

  det_finalize_kernel<<<1, 256, 0, stream>>>(partial, mined_sum,
                                             pos_cnt, mined_cnt, out);
}